// SE3HamNODEPX4_4071628997011
// MI455X (gfx1250) — compile-verified
//
#include <hip/hip_runtime.h>
#include <hip/hip_bf16.h>

typedef __attribute__((ext_vector_type(2))) float v2f;
typedef __attribute__((ext_vector_type(8))) float v8f;

#define WAVES_PB 4
#define ROWS_PB  (WAVES_PB * 16)

struct Params {
  const float* inp;
  float*       out;
  const float* w[12];
  const float* b[12];
  int nrows;
};

// CDNA5 hardware tanh (V_TANH_F32). Builtin if the toolchain has it, else
// inline asm with an embedded v_nop to satisfy the trans-op result hazard.
__device__ __forceinline__ float fast_tanh(float x) {
#if __has_builtin(__builtin_amdgcn_tanhf)
  return __builtin_amdgcn_tanhf(x);
#else
  float y;
  asm("v_tanh_f32 %0, %1\n\tv_nop" : "=v"(y) : "v"(x));
  return y;
#endif
}

// One MLP layer on a 16-row batch tile using V_WMMA_F32_16X16X4_F32.
// KP: K padded to multiple of 4 (weight rows >= real K are zero in LDS).
// NT: number of 16-wide N tiles. NACT: columns actually stored.
// Weights in LDS are K-pair interleaved: value (k, n) lives at
//   (k/2)*64 + n*2 + (k&1)   -> B fragment = one aligned ds_load_b64.
template<int KP, int NT, int NACT, bool TANH>
__device__ __forceinline__ void mlp_layer(const float* sIn, int inStride, int inCol0,
                                          const float* sW, const float* sB,
                                          float* sOut, int outStride, int lane)
{
  const int col  = lane & 15;
  const int hi   = lane >> 4;      // 0 or 1 (lane half)
  const int koff = hi << 1;        // A/B K offset for upper lane half

  v8f c[NT];
#pragma unroll
  for (int t = 0; t < NT; ++t) {
    const float bv = sB[t * 16 + col];     // same column for all 8 C VGPRs of a lane
#pragma unroll
    for (int i = 0; i < 8; ++i) c[t][i] = bv;
  }

#pragma unroll
  for (int k0 = 0; k0 < KP; k0 += 4) {
    // A fragment: lane holds row (lane&15), K = k0 + koff + {0,1}
    const float* ar = sIn + col * inStride + inCol0 + k0 + koff;
    v2f a;
    a[0] = ar[0];
    a[1] = ar[1];
#pragma unroll
    for (int t = 0; t < NT; ++t) {
      // B fragment: K = k0 + koff + {0,1}, N = t*16 + col; interleaved layout
      // => both values contiguous, 8-byte aligned.
      const v2f bb = *reinterpret_cast<const v2f*>(
          sW + ((k0 + koff) >> 1) * 64 + (t * 16 + col) * 2);
      c[t] = __builtin_amdgcn_wmma_f32_16x16x4_f32(
          /*neg_a=*/false, a, /*neg_b=*/false, bb,
          /*c_mod=*/(short)0, c[t], /*reuse_a=*/false, /*reuse_b=*/false);
    }
  }

  // C/D layout: VGPR i -> row i + 8*hi, col = t*16 + (lane&15)
#pragma unroll
  for (int t = 0; t < NT; ++t) {
    const int ccol = t * 16 + col;
#pragma unroll
    for (int i = 0; i < 8; ++i) {
      float v = c[t][i];
      if (TANH) v = fast_tanh(v);
      if (ccol < NACT) sOut[(i + (hi << 3)) * outStride + ccol] = v;
    }
  }
}

__global__ __launch_bounds__(WAVES_PB * 32)
void se3ham_kernel(Params p)
{
  // ---- LDS ----
  __shared__ __align__(16) float sWB[7040];   // 12 interleaved weight blocks + 12x32 biases
  __shared__ float sIN [WAVES_PB][16 * 24];   // staged input rows (22 used)
  __shared__ float sACT[WAVES_PB][16 * 36];   // hidden activations (32 used)
  __shared__ float sG  [WAVES_PB][16 * 24];   // g(q) output, 24 per row
  __shared__ float sLV [WAVES_PB][16 * 8];    // Dv Cholesky factors, 6 per row
  __shared__ float sLW [WAVES_PB][16 * 8];    // Dw Cholesky factors, 6 per row

  const int Ks[12]   = {12,20,20,20, 3,20,20,20, 9,20,20,20};
  const int Ns[12]   = {20,20,20,24,20,20,20, 6,20,20,20, 6};
  const int KPs[12]  = {12,20,20,20, 4,20,20,20,12,20,20,20};
  const int Woff[12] = {0,384,1024,1664,2304,2432,3072,3712,4352,4736,5376,6016};

  // Cooperative, zero-padded weight staging into K-pair-interleaved layout.
#pragma unroll
  for (int l = 0; l < 12; ++l) {
    const float* W = p.w[l];
    const int K = Ks[l], N = Ns[l], KP = KPs[l];
    float* dst = sWB + Woff[l];
    for (int f = threadIdx.x; f < KP * 32; f += blockDim.x) {
      const int k = f >> 5, nn = f & 31;
      const float v = (k < K && nn < N) ? W[k * N + nn] : 0.0f;
      dst[(k >> 1) * 64 + nn * 2 + (k & 1)] = v;
    }
    if (threadIdx.x < 32)
      sWB[6656 + l * 32 + threadIdx.x] = (threadIdx.x < N) ? p.b[l][threadIdx.x] : 0.0f;
  }
  __syncthreads();

  const int wave = threadIdx.x >> 5;
  const int lane = threadIdx.x & 31;
  const int rowBase = blockIdx.x * ROWS_PB + wave * 16;
  if (rowBase + 16 > p.nrows) return;   // after the only barrier -> safe

  float* IN  = sIN[wave];
  float* ACT = sACT[wave];
  float* G   = sG[wave];
  float* LV  = sLV[wave];
  float* LW  = sLW[wave];

  // Coalesced input staging: 16 rows x 22 f32 = 352 = 11 * 32
  {
    const float* gin = p.inp + (size_t)rowBase * 22;
#pragma unroll
    for (int j = 0; j < 11; ++j) {
      const int idx = j * 32 + lane;
      const float v = gin[idx];
      const int r  = idx / 22;
      const int cc = idx - r * 22;
      IN[r * 24 + cc] = v;
    }
  }

  const float* WB = sWB;
  const float* BB = sWB + 6656;

  // g MLP: q(12) -> 20 -> 20 -> 20 -> 24
  mlp_layer<12,2,32,true >(IN, 24, 0, WB + 0,    BB + 0*32,  ACT, 36, lane);
  mlp_layer<20,2,32,true >(ACT,36, 0, WB + 384,  BB + 1*32,  ACT, 36, lane);
  mlp_layer<20,2,32,true >(ACT,36, 0, WB + 1024, BB + 2*32,  ACT, 36, lane);
  mlp_layer<20,2,24,false>(ACT,36, 0, WB + 1664, BB + 3*32,  G,   24, lane);
  // Dv MLP: q[:3] (K=3 padded to 4) -> 20 -> 20 -> 20 -> 6
  mlp_layer< 4,2,32,true >(IN, 24, 0, WB + 2304, BB + 4*32,  ACT, 36, lane);
  mlp_layer<20,2,32,true >(ACT,36, 0, WB + 2432, BB + 5*32,  ACT, 36, lane);
  mlp_layer<20,2,32,true >(ACT,36, 0, WB + 3072, BB + 6*32,  ACT, 36, lane);
  mlp_layer<20,1, 6,false>(ACT,36, 0, WB + 3712, BB + 7*32,  LV,  8,  lane);
  // Dw MLP: q[3:12] (K=9 padded to 12, col offset 3) -> 20 -> 20 -> 20 -> 6
  mlp_layer<12,2,32,true >(IN, 24, 3, WB + 4352, BB + 8*32,  ACT, 36, lane);
  mlp_layer<20,2,32,true >(ACT,36, 0, WB + 4736, BB + 9*32,  ACT, 36, lane);
  mlp_layer<20,2,32,true >(ACT,36, 0, WB + 5376, BB + 10*32, ACT, 36, lane);
  mlp_layer<20,1, 6,false>(ACT,36, 0, WB + 6016, BB + 11*32, LW,  8,  lane);

  // ---- SE(3) port-Hamiltonian epilogue: one row per lane (lanes 0..15) ----
  if (lane < 16) {
    const float* r = IN + lane * 24;
    float R[9];
#pragma unroll
    for (int i = 0; i < 9; ++i) R[i] = r[3 + i];
    const float v0 = r[12], v1 = r[13], v2 = r[14];   // dH/dpv = v
    const float w0 = r[15], w1 = r[16], w2 = r[17];   // dH/dpw = w
    const float u0 = r[18], u1 = r[19], u2 = r[20], u3 = r[21];
    const float* g  = G  + lane * 24;
    const float* lv = LV + lane * 8;
    const float* lw = LW + lane * 8;

    const float MASSC = 1.3f;
    const float J0 = 0.012f, J1 = 0.012f, J2 = 0.02f;
    const float pv0 = MASSC * v0, pv1 = MASSC * v1, pv2 = MASSC * v2;
    const float pw0 = J0 * w0,   pw1 = J1 * w1,   pw2 = J2 * w2;

    float o[22];
    // dx = R v
    o[0] = R[0]*v0 + R[1]*v1 + R[2]*v2;
    o[1] = R[3]*v0 + R[4]*v1 + R[5]*v2;
    o[2] = R[6]*v0 + R[7]*v1 + R[8]*v2;
    // dR rows = cross(R_i, w)   (dH/dR = 0, so no extra terms)
#pragma unroll
    for (int i = 0; i < 3; ++i) {
      const float a0 = R[3*i], a1 = R[3*i+1], a2 = R[3*i+2];
      o[3 + 3*i + 0] = a1 * w2 - a2 * w1;
      o[3 + 3*i + 1] = a2 * w0 - a0 * w2;
      o[3 + 3*i + 2] = a0 * w1 - a1 * w0;
    }
    // F = g(q) (6x4) @ u
    const float F0 = g[ 0]*u0 + g[ 1]*u1 + g[ 2]*u2 + g[ 3]*u3;
    const float F1 = g[ 4]*u0 + g[ 5]*u1 + g[ 6]*u2 + g[ 7]*u3;
    const float F2 = g[ 8]*u0 + g[ 9]*u1 + g[10]*u2 + g[11]*u3;
    const float F3 = g[12]*u0 + g[13]*u1 + g[14]*u2 + g[15]*u3;
    const float F4 = g[16]*u0 + g[17]*u1 + g[18]*u2 + g[19]*u3;
    const float F5 = g[20]*u0 + g[21]*u1 + g[22]*u2 + g[23]*u3;
    // Dv v = L (L^T v),  L = [[o0,0,0],[o3,o1,0],[o4,o5,o2]]
    const float t0 = lv[0]*v0 + lv[3]*v1 + lv[4]*v2;
    const float t1 = lv[1]*v1 + lv[5]*v2;
    const float t2 = lv[2]*v2;
    const float Dv0 = lv[0]*t0;
    const float Dv1 = lv[3]*t0 + lv[1]*t1;
    const float Dv2 = lv[4]*t0 + lv[5]*t1 + lv[2]*t2;
    // Dw w
    const float s0 = lw[0]*w0 + lw[3]*w1 + lw[4]*w2;
    const float s1 = lw[1]*w1 + lw[5]*w2;
    const float s2 = lw[2]*w2;
    const float Dw0 = lw[0]*s0;
    const float Dw1 = lw[3]*s0 + lw[1]*s1;
    const float Dw2 = lw[4]*s0 + lw[5]*s1 + lw[2]*s2;

    const float MG = 1.3f * 9.81f;   // R^T dH/dx = MG * R[2,:]
    const float dpv0 = (pv1*w2 - pv2*w1) - MG * R[6] - Dv0 + F0;
    const float dpv1 = (pv2*w0 - pv0*w2) - MG * R[7] - Dv1 + F1;
    const float dpv2 = (pv0*w1 - pv1*w0) - MG * R[8] - Dv2 + F2;
    const float dpw0 = (pw1*w2 - pw2*w1) + (pv1*v2 - pv2*v1) - Dw0 + F3;
    const float dpw1 = (pw2*w0 - pw0*w2) + (pv2*v0 - pv0*v2) - Dw1 + F4;
    const float dpw2 = (pw0*w1 - pw1*w0) + (pv0*v1 - pv1*v0) - Dw2 + F5;

    const float iM  = 1.0f / 1.3f;
    const float iJ0 = 1.0f / 0.012f, iJ2 = 1.0f / 0.02f;
    o[12] = dpv0 * iM;  o[13] = dpv1 * iM;  o[14] = dpv2 * iM;
    o[15] = dpw0 * iJ0; o[16] = dpw1 * iJ0; o[17] = dpw2 * iJ2;
    o[18] = 0.0f; o[19] = 0.0f; o[20] = 0.0f; o[21] = 0.0f;

    // 22 f32 per row = 88 B -> 11 aligned float2 stores
    float2* go2 = reinterpret_cast<float2*>(p.out + (size_t)(rowBase + lane) * 22);
#pragma unroll
    for (int j = 0; j < 11; ++j) go2[j] = make_float2(o[2*j], o[2*j + 1]);
  }
}

extern "C" void kernel_launch(void* const* d_in, const int* in_sizes, int n_in,
                              void* d_out, int out_size, void* d_ws, size_t ws_size,
                              hipStream_t stream) {
  Params p;
  p.inp = (const float*)d_in[0];      // [B, 22] f32
  p.out = (float*)d_out;              // [B, 22] f32
  // d_in[1] is the python scalar t (unused).
  for (int l = 0; l < 12; ++l) {
    p.w[l] = (const float*)d_in[2 + 2*l];
    p.b[l] = (const float*)d_in[3 + 2*l];
  }
  p.nrows = in_sizes[0] / 22;
  const int grid = (p.nrows + ROWS_PB - 1) / ROWS_PB;
  se3ham_kernel<<<grid, WAVES_PB * 32, 0, stream>>>(p);
}